// STPMLP_57732950393321
// MI455X (gfx1250) — compile-verified
//
#include <hip/hip_runtime.h>
#include <math.h>

// Problem constants from the reference
#define BSZ   4
#define LSEQ  1024
#define DMOD  256
#define DFF   512
#define MTOK  (BSZ * LSEQ)   // 4096 token rows

typedef float v2f_t __attribute__((ext_vector_type(2)));
typedef float v8f_t __attribute__((ext_vector_type(8)));

// ---------------------------------------------------------------------------
// WMMA f32 GEMM:  D[M,N] = A[M,K] @ Bt[N,K]^T + bias[N]
// One wave computes a 16 x (16*NT) strip with V_WMMA_F32_16X16X4_F32.
// NT=4 independent accumulator chains -> the matrix pipe issues back-to-back
// wmma without the C-operand RAW stall, and each A fragment is reused 4x.
// Fragment layouts (ISA 7.12.2, 32-bit):
//   A (16x4): M = lane&15, K-pair = lane>>4, 2 consecutive K per lane (b64 load)
//   B (4x16): N = lane&15, K-pair = lane>>4, built from row-major (N,K) weights
//   C/D (16x16): 8 VGPRs, N = lane&15, M = v + 8*(lane>>4)
// ---------------------------------------------------------------------------
template <int Mdim, int Ndim, int Kdim, int NT>
__global__ __launch_bounds__(128)
void stp_wmma_gemm_bias(const float* __restrict__ A,
                        const float* __restrict__ Bt,   // (Ndim, Kdim) row-major
                        const float* __restrict__ bias, // (Ndim)
                        float* __restrict__ D) {
  const int wave = blockIdx.x * (blockDim.x >> 5) + (threadIdx.x >> 5);
  const int lane = threadIdx.x & 31;
  constexpr int n_groups = Ndim / (16 * NT);
  const int tm = wave / n_groups;
  const int tg = wave % n_groups;
  if (tm >= Mdim / 16) return;

  const int m0   = tm * 16;
  const int n0   = tg * 16 * NT;
  const int row  = lane & 15;
  const int half = lane >> 4;

  const float* arow = A + (size_t)(m0 + row) * Kdim + 2 * half;
  const float* brow[NT];
#pragma unroll
  for (int nt = 0; nt < NT; ++nt)
    brow[nt] = Bt + (size_t)(n0 + nt * 16 + row) * Kdim + 2 * half;

  v8f_t c[NT];
#pragma unroll
  for (int nt = 0; nt < NT; ++nt) c[nt] = (v8f_t){};

#pragma unroll 4
  for (int k0 = 0; k0 < Kdim; k0 += 4) {
    const v2f_t a = *(const v2f_t*)(arow + k0);
#pragma unroll
    for (int nt = 0; nt < NT; ++nt) {
      const v2f_t b = *(const v2f_t*)(brow[nt] + k0);
      // D = A x B + C  (f32, 16x16x4); NT independent accumulation chains
      c[nt] = __builtin_amdgcn_wmma_f32_16x16x4_f32(false, a, false, b,
                                                    (short)0, c[nt], false, false);
    }
  }

#pragma unroll
  for (int nt = 0; nt < NT; ++nt) {
    const int n = n0 + nt * 16 + row;
    const float bn = bias[n];
#pragma unroll
    for (int v = 0; v < 8; ++v) {
      const int m = m0 + v + 8 * half;
      D[(size_t)m * Ndim + n] = c[nt][v] + bn;
    }
  }
}

// ---------------------------------------------------------------------------
// Fast-weight scan. One wave32 per (batch, dff-row): 2048 independent waves.
// Each lane owns 8 contiguous columns of the F row, held in VGPRs for the
// whole sequence (zero state memory traffic). Software-pipelined: x_{t+1} and
// h_static_{t+1} loads issue before step t's ds-butterfly + erf, hiding their
// L2 latency off the recurrence critical path. GELU(h) is stored so the
// output GEMM is a plain WMMA GEMM.
// ---------------------------------------------------------------------------
__global__ __launch_bounds__(256)
void stp_scan_kernel(const float* __restrict__ x,        // (B, L, D)
                     const float* __restrict__ Hs,       // (B*L, DFF) = x@W1^T+b1
                     const float* __restrict__ Lambda,   // (DFF, D)
                     const float* __restrict__ Gamma,    // (DFF, D)
                     float* __restrict__ Hg) {           // (B*L, DFF) gelu(h_pre)
  const int wave = blockIdx.x * (blockDim.x >> 5) + (threadIdx.x >> 5);
  const int lane = threadIdx.x & 31;
  if (wave >= BSZ * DFF) return;
  const int b  = wave >> 9;        // / DFF
  const int i  = wave & (DFF - 1); // % DFF
  const int jb = lane * 8;         // this lane's 8 columns

  // Per-lane constants: retention r = 1 - sigmoid(Lambda), gamma g
  float r[8], g[8], F[8];
  const float* lam = Lambda + (size_t)i * DMOD + jb;
  const float* gam = Gamma  + (size_t)i * DMOD + jb;
#pragma unroll
  for (int u = 0; u < 8; ++u) {
    const float s = 1.0f / (1.0f + __expf(-lam[u]));
    r[u] = 1.0f - s;
    g[u] = gam[u];
    F[u] = 0.0f;
  }

  const float* xb     = x  + (size_t)b * LSEQ * DMOD + jb;
  const float* hs_ptr = Hs + (size_t)b * LSEQ * DFF + i;
  float*       hg_ptr = Hg + (size_t)b * LSEQ * DFF + i;

  // Pipeline prologue: step-0 operands
  float4 cx0 = *(const float4*)(xb);
  float4 cx1 = *(const float4*)(xb + 4);
  float  chs = hs_ptr[0];

  for (int t = 0; t < LSEQ; ++t) {
    // Issue t+1 loads first; latency overlaps the reduction chain below.
    float4 nx0, nx1;
    float  nhs = 0.0f;
    if (t + 1 < LSEQ) {
      const float* xn = xb + (size_t)(t + 1) * DMOD;
      nx0 = *(const float4*)(xn);
      nx1 = *(const float4*)(xn + 4);
      nhs = hs_ptr[(size_t)(t + 1) * DFF];
      if (t + 2 < LSEQ) __builtin_prefetch(xn + DMOD, 0, 1);
    }

    const float xv[8] = {cx0.x, cx0.y, cx0.z, cx0.w, cx1.x, cx1.y, cx1.z, cx1.w};

    // dot(F_row, x_t): per-lane partial + wave32 xor-butterfly reduction
    float p = 0.0f;
#pragma unroll
    for (int u = 0; u < 8; ++u) p = fmaf(F[u], xv[u], p);
#pragma unroll
    for (int off = 16; off >= 1; off >>= 1) p += __shfl_xor(p, off, 32);

    const float h = chs + p;   // h_pre, identical in all lanes

    // exact GELU (erf form), one writer per wave
    const float gh = 0.5f * h * (1.0f + erff(h * 0.70710678118654752f));
    if (lane == 0) hg_ptr[(size_t)t * DFF] = gh;

    // F <- r*F + g * h * x_t   (pre-update F was used for the dot, as in ref)
#pragma unroll
    for (int u = 0; u < 8; ++u) F[u] = fmaf(r[u], F[u], g[u] * h * xv[u]);

    cx0 = nx0; cx1 = nx1; chs = nhs;
  }
}

// ---------------------------------------------------------------------------
// Launcher: GEMM1 (H_static) -> scan (GELU(h), F in VGPRs) -> GEMM2 (out)
// Inputs: 0:x 1:W1 2:b1 3:W2 4:b2 5:Lambda_raw 6:Gamma
// ---------------------------------------------------------------------------
extern "C" void kernel_launch(void* const* d_in, const int* in_sizes, int n_in,
                              void* d_out, int out_size, void* d_ws, size_t ws_size,
                              hipStream_t stream) {
  const float* x   = (const float*)d_in[0];
  const float* W1  = (const float*)d_in[1];
  const float* b1  = (const float*)d_in[2];
  const float* W2  = (const float*)d_in[3];
  const float* b2  = (const float*)d_in[4];
  const float* Lam = (const float*)d_in[5];
  const float* Gam = (const float*)d_in[6];
  float* out = (float*)d_out;

  float* Hs = (float*)d_ws;                       // (4096, 512) = 8 MB
  float* Hg = Hs + (size_t)MTOK * DFF;            // (4096, 512) = 8 MB

  // GEMM1: H_static = x @ W1^T + b1   (M=4096, N=512, K=256), 4 N-tiles/wave
  {
    constexpr int NT = 4;
    const int waves = (MTOK / 16) * (DFF / (16 * NT));   // 2048
    stp_wmma_gemm_bias<MTOK, DFF, DMOD, NT>
        <<<waves / 4, 128, 0, stream>>>(x, W1, b1, Hs);
  }

  // Scan: 2048 waves (one per F row), F entirely in VGPRs
  stp_scan_kernel<<<(BSZ * DFF) / 8, 256, 0, stream>>>(x, Hs, Lam, Gam, Hg);

  // GEMM2: out = GELU(H) @ W2^T + b2  (M=4096, N=256, K=512), 4 N-tiles/wave
  {
    constexpr int NT = 4;
    const int waves = (MTOK / 16) * (DMOD / (16 * NT));  // 1024
    stp_wmma_gemm_bias<MTOK, DMOD, DFF, NT>
        <<<waves / 4, 128, 0, stream>>>(Hg, W2, b2, out);
  }
}